// Scalar_1881195675680
// MI455X (gfx1250) — compile-verified
//
#include <hip/hip_runtime.h>
#include <hip/hip_bf16.h>

// ---------------------------------------------------------------------------
// 2-D scalar wave propagator with CPML, 4th-order FD, fused per-timestep step.
// Grid: padded NPAD x NPAD (pad=2 each side, pads always exactly 0).
// One fused kernel per timestep (64 launches); everything stays L2-resident
// (~42 MB working set vs 192 MB L2).  wfc tile staged into LDS via CDNA5
// async global->LDS B128 loads with s_wait_asynccnt.  Receiver sampling is
// folded into the step kernel as a spare blockIdx.z slice (read-only on wfc).
// ---------------------------------------------------------------------------

#define NPAD   404
#define NN     400
#define BATCH  8
#define NTS    64
#define NSRCS  4
#define NRECS  128
#define TX     32
#define TY     32

#define PLANE ((size_t)NPAD * (size_t)NPAD)

// 4th-order FD coefficients
#define FD_C1 (8.0f/12.0f)      // first derivative +-1
#define FD_C2 (-1.0f/12.0f)     // first derivative +-2
#define FD_D1 (4.0f/3.0f)       // second derivative +-1
#define FD_D2 (-1.0f/12.0f)     // second derivative +-2
#define ODX   0.25f             // 1/DX
#define ODX2  0.0625f           // 1/DX^2

#if defined(__HIP_DEVICE_COMPILE__) && \
    __has_builtin(__builtin_amdgcn_global_load_async_to_lds_b128) && \
    __has_builtin(__builtin_amdgcn_s_wait_asynccnt)
#define USE_ASYNC_LDS 1
#else
#define USE_ASYNC_LDS 0
#endif

#if USE_ASYNC_LDS
typedef int v4i_ __attribute__((vector_size(16)));
// AS attribute in the typedef applies to the pointee: pointer to AS1/AS3 v4i.
typedef __attribute__((address_space(1))) v4i_* gv4i_ptr;
typedef __attribute__((address_space(3))) v4i_* lv4i_ptr;
#endif

// ---------------------------------------------------------------------------
// Fused timestep.  One 32x32 output tile per workgroup, block = (32,8).
//   LDS: s_wfc  [40][48]  wfc with halo 4 (x) / staged as aligned B128 chunks (y)
//        s_psix [36][32]  NEW psix with x-halo 2 (recomputed in halo)
//        s_psiy [32][36]  NEW psiy with y-halo 2
// In-place: wfn overwrites wfp (pointwise in wfp); zetax/zetay in place.
// psix/psiy double-buffered (halo recompute reads neighbors' OLD psi).
// blockIdx.z == BATCH: spare slice records receivers from wfc (read-only).
__global__ __launch_bounds__(256) void step_kernel(
    const float* __restrict__ wfc,      float* __restrict__ wfp,
    const float* __restrict__ psix_old, float* __restrict__ psix_new,
    const float* __restrict__ psiy_old, float* __restrict__ psiy_new,
    float* __restrict__ zetax, float* __restrict__ zetay,
    const float* __restrict__ v2dt2,
    const float* __restrict__ ax, const float* __restrict__ ay,
    const float* __restrict__ bx, const float* __restrict__ by,
    const int* __restrict__ rloc, float* __restrict__ rec_out, int tstep) {

    __shared__ __align__(16) float s_wfc[40][48];
    __shared__ float s_psix[36][32];
    __shared__ float s_psiy[32][36];

    const int t = threadIdx.y * 32 + threadIdx.x;

    if (blockIdx.z == BATCH) {
        // -------- receiver recording slice (reads wfc only) ----------------
        const int sb = blockIdx.y * gridDim.x + blockIdx.x;
        const int idx = sb * 256 + t;
        if (idx < BATCH * NRECS) {
            const int b = idx / NRECS, r = idx % NRECS;
            const int rx = rloc[(b * NRECS + r) * 2 + 0] + 2;
            const int ry = rloc[(b * NRECS + r) * 2 + 1] + 2;
            rec_out[(size_t)(b * NRECS + r) * NTS + tstep] =
                wfc[(size_t)b * PLANE + (size_t)rx * NPAD + ry];
        }
        return;
    }

    const int b  = blockIdx.z;
    const int x0 = 2 + blockIdx.y * TX;   // first interior row of tile
    const int y0 = 2 + blockIdx.x * TY;   // first interior col of tile

    const float* wfc_b = wfc + (size_t)b * PLANE;

    // ---- Phase 1: stage wfc rows [x0-4, x0+36) x cols [y0-6, y0+42) --------
    // Column window starts at y0-6 so every 4-float chunk is 16B aligned and
    // either fully inside [0, NPAD) or fully outside (NPAD % 4 == 0).
    for (int k = t; k < 40 * 12; k += 256) {
        const int r  = k / 12;
        const int cc = k % 12;
        const int gx = x0 - 4 + r;
        const int gy = y0 - 6 + cc * 4;
        if (gx >= 0 && gx < NPAD && gy >= 0 && gy < NPAD) {
#if USE_ASYNC_LDS
            __builtin_amdgcn_global_load_async_to_lds_b128(
                (gv4i_ptr)(wfc_b + (size_t)gx * NPAD + gy),
                (lv4i_ptr)&s_wfc[r][cc * 4],
                0, 0);
#else
            const float4 v = *(const float4*)(wfc_b + (size_t)gx * NPAD + gy);
            *(float4*)&s_wfc[r][cc * 4] = v;
#endif
        } else {
            s_wfc[r][cc * 4 + 0] = 0.0f;
            s_wfc[r][cc * 4 + 1] = 0.0f;
            s_wfc[r][cc * 4 + 2] = 0.0f;
            s_wfc[r][cc * 4 + 3] = 0.0f;
        }
    }
#if USE_ASYNC_LDS
    __builtin_amdgcn_s_wait_asynccnt(0);
#endif
    __syncthreads();

    // ---- Phase 2: new psix on [x0-2, x0+34) x [y0, y0+32)  (0 outside interior)
    for (int k = t; k < 36 * 32; k += 256) {
        const int r  = k >> 5;          // 0..35
        const int c  = k & 31;          // 0..31
        const int gx = x0 - 2 + r;
        const int gy = y0 + c;
        float val = 0.0f;
        if (gx >= 2 && gx <= NPAD - 3 && gy >= 2 && gy <= NPAD - 3) {
            const int wr = r + 2, wcc = c + 6;      // index into s_wfc
            const float dwdx = (FD_C1 * (s_wfc[wr + 1][wcc] - s_wfc[wr - 1][wcc]) +
                                FD_C2 * (s_wfc[wr + 2][wcc] - s_wfc[wr - 2][wcc])) * ODX;
            val = ax[gx] * psix_old[(size_t)b * PLANE + (size_t)gx * NPAD + gy] +
                  bx[gx] * dwdx;
        }
        s_psix[r][c] = val;
    }

    // ---- Phase 3: new psiy on [x0, x0+32) x [y0-2, y0+34) ------------------
    for (int k = t; k < 32 * 36; k += 256) {
        const int r  = k / 36;          // 0..31
        const int c  = k % 36;          // 0..35
        const int gx = x0 + r;
        const int gy = y0 - 2 + c;
        float val = 0.0f;
        if (gx >= 2 && gx <= NPAD - 3 && gy >= 2 && gy <= NPAD - 3) {
            const int wr = r + 4, wcc = c + 4;
            const float dwdy = (FD_C1 * (s_wfc[wr][wcc + 1] - s_wfc[wr][wcc - 1]) +
                                FD_C2 * (s_wfc[wr][wcc + 2] - s_wfc[wr][wcc - 2])) * ODX;
            val = ay[gy] * psiy_old[(size_t)b * PLANE + (size_t)gx * NPAD + gy] +
                  by[gy] * dwdy;
        }
        s_psiy[r][c] = val;
    }
    __syncthreads();

    // ---- Phase 4: zetax/zetay/wfn + store all outputs ----------------------
    for (int i = 0; i < 4; ++i) {
        const int r  = threadIdx.y + 8 * i;   // 0..31
        const int c  = threadIdx.x;           // 0..31
        const int gx = x0 + r;
        const int gy = y0 + c;
        if (gx > NPAD - 3 || gy > NPAD - 3) continue;

        const int wr = r + 4, wcc = c + 6;
        const size_t g = (size_t)b * PLANE + (size_t)gx * NPAD + gy;

        const float w0 = s_wfc[wr][wcc];
        const float d2x = (-2.5f * w0 +
                           FD_D1 * (s_wfc[wr + 1][wcc] + s_wfc[wr - 1][wcc]) +
                           FD_D2 * (s_wfc[wr + 2][wcc] + s_wfc[wr - 2][wcc])) * ODX2;
        const float d2y = (-2.5f * w0 +
                           FD_D1 * (s_wfc[wr][wcc + 1] + s_wfc[wr][wcc - 1]) +
                           FD_D2 * (s_wfc[wr][wcc + 2] + s_wfc[wr][wcc - 2])) * ODX2;

        // psix_x = diff_x(psix_new); s_psix row r' maps to gx' = x0-2+r'
        const float psix_x = (FD_C1 * (s_psix[r + 3][c] - s_psix[r + 1][c]) +
                              FD_C2 * (s_psix[r + 4][c] - s_psix[r][c])) * ODX;
        const float psiy_y = (FD_C1 * (s_psiy[r][c + 3] - s_psiy[r][c + 1]) +
                              FD_C2 * (s_psiy[r][c + 4] - s_psiy[r][c])) * ODX;

        const float axv = ax[gx], bxv = bx[gx];
        const float ayv = ay[gy], byv = by[gy];

        const float ztx = axv * zetax[g] + bxv * (d2x + psix_x);
        const float zty = ayv * zetay[g] + byv * (d2y + psiy_y);
        const float wfn = v2dt2[(size_t)gx * NPAD + gy] *
                              (d2x + d2y + psix_x + psiy_y + ztx + zty) +
                          2.0f * w0 - wfp[g];

        zetax[g]    = ztx;
        zetay[g]    = zty;
        wfp[g]      = wfn;              // wfn overwrites old wfp (pointwise)
        psix_new[g] = s_psix[r + 2][c];
        psiy_new[g] = s_psiy[r][c + 2];
    }
}

// ---------------------------------------------------------------------------
// v2dt2 = ((1500 + 1000*v_pad) * DT)^2,  shared across batch
__global__ __launch_bounds__(256) void v2_kernel(const float* __restrict__ v_pad,
                                                 float* __restrict__ v2) {
    int i = blockIdx.x * 256 + threadIdx.x;
    if (i >= (int)PLANE) return;
    float s = (1500.0f + 1000.0f * v_pad[i]) * 4e-4f;
    v2[i] = s * s;
}

// ---------------------------------------------------------------------------
// Initialize all 8 padded state arrays: interiors from inputs, pads = 0.
__global__ __launch_bounds__(256) void init_kernel(
    float* __restrict__ wfA, float* __restrict__ wfB,
    float* __restrict__ pxA, float* __restrict__ pxB,
    float* __restrict__ pyA, float* __restrict__ pyB,
    float* __restrict__ zx,  float* __restrict__ zy,
    const float* __restrict__ wf0, const float* __restrict__ wfm1,
    const float* __restrict__ px0, const float* __restrict__ py0,
    const float* __restrict__ zx0, const float* __restrict__ zy0) {
    size_t i = (size_t)blockIdx.x * 256 + threadIdx.x;
    if (i >= (size_t)BATCH * PLANE) return;
    int b   = (int)(i / PLANE);
    int rem = (int)(i % PLANE);
    int x = rem / NPAD, y = rem % NPAD;
    bool in = (x >= 2 && x < 2 + NN && y >= 2 && y < 2 + NN);
    size_t j = in ? ((size_t)b * NN * NN + (size_t)(x - 2) * NN + (y - 2)) : 0;
    wfA[i] = in ? wf0[j]  : 0.0f;
    wfB[i] = in ? wfm1[j] : 0.0f;
    pxA[i] = in ? px0[j]  : 0.0f;
    pyA[i] = in ? py0[j]  : 0.0f;
    zx[i]  = in ? zx0[j]  : 0.0f;
    zy[i]  = in ? zy0[j]  : 0.0f;
    pxB[i] = 0.0f;
    pyB[i] = 0.0f;
}

// ---------------------------------------------------------------------------
// Inject sources into the freshly computed wavefield (atomic: dup locations).
__global__ void src_kernel(float* __restrict__ wfn,
                           const float* __restrict__ amps,
                           const int* __restrict__ sloc, int t) {
    int idx = threadIdx.x;
    if (idx >= BATCH * NSRCS) return;
    int b = idx / NSRCS, s = idx % NSRCS;
    int sx = sloc[(b * NSRCS + s) * 2 + 0] + 2;
    int sy = sloc[(b * NSRCS + s) * 2 + 1] + 2;
    atomicAdd(&wfn[(size_t)b * PLANE + (size_t)sx * NPAD + sy],
              amps[(size_t)(b * NTS + t) * NSRCS + s]);
}

// ---------------------------------------------------------------------------
// Crop the six padded state arrays into d_out (return order of the reference).
__global__ __launch_bounds__(256) void finalize_kernel(
    float* __restrict__ out,
    const float* __restrict__ wfc, const float* __restrict__ wfp,
    const float* __restrict__ px,  const float* __restrict__ py,
    const float* __restrict__ zx,  const float* __restrict__ zy) {
    size_t i = (size_t)blockIdx.x * 256 + threadIdx.x;
    const size_t per = (size_t)BATCH * NN * NN;
    if (i >= 6 * per) return;
    int which  = (int)(i / per);
    size_t rem = i % per;
    int b  = (int)(rem / ((size_t)NN * NN));
    int r2 = (int)(rem % ((size_t)NN * NN));
    int x = r2 / NN, y = r2 % NN;
    const float* s = wfc;
    if      (which == 1) s = wfp;
    else if (which == 2) s = px;
    else if (which == 3) s = py;
    else if (which == 4) s = zx;
    else if (which == 5) s = zy;
    out[i] = s[(size_t)b * PLANE + (size_t)(x + 2) * NPAD + (y + 2)];
}

// ---------------------------------------------------------------------------
extern "C" void kernel_launch(void* const* d_in, const int* in_sizes, int n_in,
                              void* d_out, int out_size, void* d_ws, size_t ws_size,
                              hipStream_t stream) {
    const float* v_pad = (const float*)d_in[0];
    const float* samp  = (const float*)d_in[1];
    const float* wf0   = (const float*)d_in[2];
    const float* wfm1  = (const float*)d_in[3];
    const float* px0   = (const float*)d_in[4];
    const float* py0   = (const float*)d_in[5];
    const float* zx0   = (const float*)d_in[6];
    const float* zy0   = (const float*)d_in[7];
    const float* ax    = (const float*)d_in[8];
    const float* ay    = (const float*)d_in[9];
    const float* bx    = (const float*)d_in[10];
    const float* by    = (const float*)d_in[11];
    const int*   sloc  = (const int*)d_in[12];
    const int*   rloc  = (const int*)d_in[13];

    // Workspace layout (floats), each array 16B aligned (PLANE % 4 == 0).
    float* ws  = (float*)d_ws;
    float* v2  = ws;                         // PLANE
    float* wfA = v2  + PLANE;                // BATCH*PLANE each
    float* wfB = wfA + (size_t)BATCH * PLANE;
    float* pxA = wfB + (size_t)BATCH * PLANE;
    float* pxB = pxA + (size_t)BATCH * PLANE;
    float* pyA = pxB + (size_t)BATCH * PLANE;
    float* pyB = pyA + (size_t)BATCH * PLANE;
    float* zx  = pyB + (size_t)BATCH * PLANE;
    float* zy  = zx  + (size_t)BATCH * PLANE;

    v2_kernel<<<(int)((PLANE + 255) / 256), 256, 0, stream>>>(v_pad, v2);
    init_kernel<<<(int)((BATCH * PLANE + 255) / 256), 256, 0, stream>>>(
        wfA, wfB, pxA, pxB, pyA, pyB, zx, zy, wf0, wfm1, px0, py0, zx0, zy0);

    float* out = (float*)d_out;
    float* rec = out + (size_t)6 * BATCH * NN * NN;

    float* wf_c = wfA; float* wf_p = wfB;
    float* px_c = pxA; float* px_n = pxB;
    float* py_c = pyA; float* py_n = pyB;

    // z = BATCH+1: last z-slice records receivers (read-only on wfc).
    const dim3 sgrid((NN + TY - 1) / TY, (NN + TX - 1) / TX, BATCH + 1);
    const dim3 sblock(32, 8);

    for (int tstep = 0; tstep < NTS; ++tstep) {
        step_kernel<<<sgrid, sblock, 0, stream>>>(wf_c, wf_p, px_c, px_n,
                                                  py_c, py_n, zx, zy, v2,
                                                  ax, ay, bx, by,
                                                  rloc, rec, tstep);
        src_kernel<<<1, BATCH * NSRCS, 0, stream>>>(wf_p, samp, sloc, tstep);
        float* tmp;
        tmp = wf_c; wf_c = wf_p; wf_p = tmp;
        tmp = px_c; px_c = px_n; px_n = tmp;
        tmp = py_c; py_c = py_n; py_n = tmp;
    }

    finalize_kernel<<<(int)(((size_t)6 * BATCH * NN * NN + 255) / 256), 256, 0,
                      stream>>>(out, wf_c, wf_p, px_c, py_c, zx, zy);
}